// CurrentHistoricalAttention_57784490000884
// MI455X (gfx1250) — compile-verified
//
#include <hip/hip_runtime.h>
#include <hip/hip_bf16.h>

typedef __attribute__((ext_vector_type(2))) float v2f;
typedef __attribute__((ext_vector_type(8))) float v8f;
typedef __attribute__((ext_vector_type(4))) unsigned int u32x4;
typedef __attribute__((ext_vector_type(8))) int i32x8;
typedef __attribute__((ext_vector_type(4))) int i32x4;

constexpr int kH = 512;   // hidden
constexpr int kB = 1024;  // batch
constexpr int kD = 256;   // history

// LDS tile geometry for K2: 16 rows x 512 f32, TDM-padded 2 dwords every 256
// => effective row stride 516 words (banks 4m / 4m+2 per half-wave, conflict-free)
constexpr int kRowStride = 516;                 // words
constexpr int kWaveLdsWords = 8272;             // >= 16*516+2, 16B aligned
constexpr int kWavesK2 = 4;                     // 132352 B dynamic LDS / block

// ---------------------------------------------------------------------------
// fast tanh: tanh(x) = sign(x) * (1 - e^{-2|x|}) / (1 + e^{-2|x|})
// ---------------------------------------------------------------------------
__device__ __forceinline__ float fast_tanhf(float x) {
  float ax = __builtin_fabsf(x);
  float e  = __expf(-2.0f * ax);
  float r  = (1.0f - e) / (1.0f + e);
  return __builtin_copysignf(r, x);
}

// CDNA5 f32 WMMA (16x16x4, D = A*B + C)
__device__ __forceinline__ v8f wmma_f32_4(v2f a, v2f b, v8f c) {
  return __builtin_amdgcn_wmma_f32_16x16x4_f32(false, a, false, b, (short)0, c,
                                               false, false);
}

// ---------------------------------------------------------------------------
// TDM: DMA one 16x512 f32 tile (row stride = kH in memory) into LDS at
// lds_off, inserting 2 pad dwords every 256 dwords (ISA §8 D# layout).
// Group0: count=1 | lds_addr | global_addr[56:0] | type=2
// Group1: data_size=4B, pad_enable, pad_interval=256dw(code7),
//         pad_amount=2dw(code1); dims 512x16, tile 512x16, stride0=kH
// Toolchain is the 6-arg variant: (g0, g1, g2, g3, g4, cpol).
// ---------------------------------------------------------------------------
__device__ __forceinline__ void tdm_load_tile_16x512(unsigned int lds_off,
                                                     const float* gptr) {
  unsigned long long ga = (unsigned long long)(uintptr_t)gptr;

  u32x4 g0;
  g0.x = 1u;                                            // count=1, user mode
  g0.y = lds_off;                                       // LDS byte address
  g0.z = (unsigned int)ga;                              // global addr [31:0]
  g0.w = (unsigned int)((ga >> 32) & 0x01FFFFFFu) | (2u << 30);  // [56:32]|type

  const unsigned int td0 = 512u, td1 = 16u, tile0 = 512u, tile1 = 16u;
  const unsigned long long s0 = (unsigned long long)kH;  // row stride (elems)

  i32x8 g1;
  g1[0] = (int)((2u << 16) | (1u << 20) | (7u << 22) | (1u << 25));
  g1[1] = (int)((td0 & 0xFFFFu) << 16);                 // abar=0 | td0.lo
  g1[2] = (int)((td0 >> 16) | ((td1 & 0xFFFFu) << 16)); // td0.hi | td1.lo
  g1[3] = (int)((td1 >> 16) | (tile0 << 16));           // td1.hi | tile_dim0
  g1[4] = (int)(tile1);                                 // tile_dim1 | tile_dim2=0
  g1[5] = (int)(s0 & 0xFFFFFFFFu);                      // stride0[31:0]
  g1[6] = (int)((unsigned int)(s0 >> 32));              // stride0[47:32]|s1.lo=0
  g1[7] = 0;                                            // stride1[47:16]=0

  i32x4 z4 = {0, 0, 0, 0};                              // groups 2/3 unused (2D)
  i32x8 z8 = {0, 0, 0, 0, 0, 0, 0, 0};                  // extra group (unused)
  __builtin_amdgcn_tensor_load_to_lds(g0, g1, z4, z4, z8, 0);
}

// ---------------------------------------------------------------------------
// K1: q_proj[b, n] = sum_k concat(h_tilde, c_t)[b, k] * Waq[n, k] + ba[n]
// One wave = one 16x16 tile; K loop over 2H in steps of 4 (WMMA f32 16x16x4).
// ---------------------------------------------------------------------------
__global__ __launch_bounds__(256) void k1_qproj(
    const float* __restrict__ h_tilde, const float* __restrict__ c_t,
    const float* __restrict__ Waq, const float* __restrict__ ba,
    float* __restrict__ q_ws) {
  const int lane = threadIdx.x & 31;
  const int wave = threadIdx.x >> 5;
  const int tile = blockIdx.x * 8 + wave;  // (kB/16)*(kH/16) = 2048 tiles
  const int b0 = (tile >> 5) * 16;
  const int n0 = (tile & 31) * 16;
  const int m  = lane & 15;
  const int kh = (lane >> 4) * 2;
  const int K2H = 2 * kH;

  v8f acc = {};
  for (int k0 = 0; k0 < K2H; k0 += 4) {
    const int kk = k0 + kh;  // even; kk,kk+1 stay within the same half
    const float* asrc = (kk < kH)
                            ? (h_tilde + (size_t)(b0 + m) * kH + kk)
                            : (c_t + (size_t)(b0 + m) * kH + (kk - kH));
    v2f a  = *(const v2f*)asrc;
    v2f bm = *(const v2f*)(Waq + (size_t)(n0 + m) * K2H + kk);
    acc = wmma_f32_4(a, bm, acc);
  }

  const int   n    = n0 + m;
  const float bias = ba[n];
  const int mbase  = b0 + (lane >> 4) * 8;
#pragma unroll
  for (int r = 0; r < 8; ++r)
    q_ws[(size_t)(mbase + r) * kH + n] = acc[r] + bias;
}

// ---------------------------------------------------------------------------
// K2: s[b,d] = sum_n tanh( (h_history[b,d,:] @ Wah[n,:]) + q_proj[b,n] ) * v_t[n]
// Per wave: TDM-stage its 16x512 A tile into LDS (coalesced DMA replaces a
// 2KB-strided 16-line gather), s_wait_tensorcnt, then 32 N-tiles x 128
// v_wmma_f32_16x16x4_f32 with A from LDS (bank-conflict-free via TDM pad)
// and B (L2-resident Wah) from global.
// ---------------------------------------------------------------------------
__global__ __launch_bounds__(kWavesK2 * 32) void k2_scores(
    const float* __restrict__ hh, const float* __restrict__ Wah,
    const float* __restrict__ q_ws, const float* __restrict__ v_t,
    float* __restrict__ s_ws) {
  extern __shared__ float smem[];
  const int lane = threadIdx.x & 31;
  const int wave = threadIdx.x >> 5;
  const int b  = blockIdx.x;
  const int d0 = (blockIdx.y * kWavesK2 + wave) * 16;
  const int m  = lane & 15;
  const int kh = (lane >> 4) * 2;

  float* As = smem + (size_t)wave * kWaveLdsWords;
  const float* gtile = hh + ((size_t)b * kD + d0) * kH;

  tdm_load_tile_16x512((unsigned int)(uintptr_t)As, gtile);
  __builtin_amdgcn_s_wait_tensorcnt(0);  // TENSORcnt is per-wave: no barrier

  const float* arow = As + kRowStride * m;  // + kc + 2*(kc>>8)

  float srow[8];
#pragma unroll
  for (int r = 0; r < 8; ++r) srow[r] = 0.0f;

  for (int nt = 0; nt < 32; ++nt) {
    const int n0 = nt * 16;
    const float* brow = Wah + (size_t)(n0 + m) * kH + kh;

    v8f acc = {};
#pragma unroll 8
    for (int k0 = 0; k0 < kH; k0 += 4) {
      const int kc = kh + k0;
      v2f a  = *(const v2f*)(arow + kc + 2 * (kc >> 8));  // even idx: 8B aligned
      v2f bm = *(const v2f*)(brow + k0);
      acc = wmma_f32_4(a, bm, acc);
    }

    const int   n  = n0 + m;
    const float qb = q_ws[(size_t)b * kH + n];  // already includes ba
    const float vt = v_t[n];
#pragma unroll
    for (int r = 0; r < 8; ++r) srow[r] += fast_tanhf(acc[r] + qb) * vt;
  }

  // reduce over the 16 n-lanes within each half-wave
#pragma unroll
  for (int r = 0; r < 8; ++r) {
    float v = srow[r];
    v += __shfl_xor(v, 1, 32);
    v += __shfl_xor(v, 2, 32);
    v += __shfl_xor(v, 4, 32);
    v += __shfl_xor(v, 8, 32);
    srow[r] = v;
  }
  if (m == 0) {
    const int dd = d0 + (lane >> 4) * 8;
#pragma unroll
    for (int r = 0; r < 8; ++r) s_ws[(size_t)b * kD + dd + r] = srow[r];
  }
}

// ---------------------------------------------------------------------------
// K3: alpha[b, :] = softmax(s[b, :]) over D = 256
// ---------------------------------------------------------------------------
__global__ __launch_bounds__(256) void k3_softmax(
    const float* __restrict__ s_ws, float* __restrict__ alpha) {
  __shared__ float red[kD];
  const int b = blockIdx.x;
  const int t = threadIdx.x;  // == d

  const float x = s_ws[(size_t)b * kD + t];
  red[t] = x;
  __syncthreads();
  for (int off = kD / 2; off > 0; off >>= 1) {
    if (t < off) red[t] = fmaxf(red[t], red[t + off]);
    __syncthreads();
  }
  const float mx = red[0];
  __syncthreads();

  const float e = __expf(x - mx);
  red[t] = e;
  __syncthreads();
  for (int off = kD / 2; off > 0; off >>= 1) {
    if (t < off) red[t] = red[t] + red[t + off];
    __syncthreads();
  }
  alpha[(size_t)b * kD + t] = e * (1.0f / red[0]);
}

// ---------------------------------------------------------------------------
// K4: e_t[b, h] = sum_d alpha[b, d] * h_history[b, d, h]
// Coalesced over h; alpha loads are block-uniform -> scalar loads.
// ---------------------------------------------------------------------------
__global__ __launch_bounds__(128) void k4_context(
    const float* __restrict__ hh, const float* __restrict__ alpha,
    float* __restrict__ e_t) {
  const int b = blockIdx.x;
  const int h = blockIdx.y * 128 + threadIdx.x;
  const float* base = hh + ((size_t)b * kD) * kH + h;
  const float* al   = alpha + (size_t)b * kD;

  float acc = 0.0f;
#pragma unroll 4
  for (int d = 0; d < kD; ++d) acc = fmaf(al[d], base[(size_t)d * kH], acc);
  e_t[(size_t)b * kH + h] = acc;
}

// ---------------------------------------------------------------------------
extern "C" void kernel_launch(void* const* d_in, const int* in_sizes, int n_in,
                              void* d_out, int out_size, void* d_ws,
                              size_t ws_size, hipStream_t stream) {
  const float* h_tilde = (const float*)d_in[0];  // [B, H]
  const float* c_t     = (const float*)d_in[1];  // [B, H]
  const float* hh      = (const float*)d_in[2];  // [B, D, H]
  const float* Waq     = (const float*)d_in[3];  // [H, 2H]
  const float* Wah     = (const float*)d_in[4];  // [H, H]
  const float* ba      = (const float*)d_in[5];  // [H]
  const float* v_t     = (const float*)d_in[6];  // [H]

  float* out   = (float*)d_out;
  float* e_t   = out;                      // [B, H]
  float* alpha = out + (size_t)kB * kH;    // [B, D]

  float* q_ws = (float*)d_ws;              // [B, H] (includes +ba)
  float* s_ws = q_ws + (size_t)kB * kH;    // [B, D]

  k1_qproj<<<256, 256, 0, stream>>>(h_tilde, c_t, Waq, ba, q_ws);

  const size_t k2_lds = (size_t)kWavesK2 * kWaveLdsWords * sizeof(float);
  k2_scores<<<dim3(kB, kD / (kWavesK2 * 16)), kWavesK2 * 32, k2_lds, stream>>>(
      hh, Wah, q_ws, v_t, s_ws);

  k3_softmax<<<kB, kD, 0, stream>>>(s_ws, alpha);
  k4_context<<<dim3(kB, kH / 128), 128, 0, stream>>>(hh, alpha, e_t);
}